// SineGenerator_47571057770836
// MI455X (gfx1250) — compile-verified
//
#include <hip/hip_runtime.h>
#include <math.h>

typedef float v2f __attribute__((ext_vector_type(2)));
typedef float v8f __attribute__((ext_vector_type(8)));

#define NSAMP   240000
#define SRATE   24000
#define NH      9
#define TILE    256
#define CHUNK   4096              // 16 tiles per workgroup (8 waves x 2 tiles)
#define PADCOL  17                // 16 rows + 1 pad float -> conflict-free LDS columns
#define TILE_LDS (16 * PADCOL)    // 272 floats per 256-sample tile

__device__ __forceinline__ float waveReduceAdd(float v) {
#pragma unroll
  for (int d = 1; d < 32; d <<= 1) v += __shfl_xor(v, d, 32);
  return v;
}

// ---------------- Pass 1: per-block partial sums of c*f0 ----------------
__global__ __launch_bounds__(256) void sg_partials(const float* __restrict__ f0,
                                                   float* __restrict__ partials,
                                                   int N, int nblk) {
  const float C = 6.28318530717958647692f / (float)SRATE;
  int b = blockIdx.y, blk = blockIdx.x;
  const float* src = f0 + (size_t)b * N + (size_t)blk * CHUNK;
  int remain = N - blk * CHUNK;
  float s = 0.f;
#pragma unroll
  for (int i = 0; i < CHUNK / 256; ++i) {
    int idx = threadIdx.x + 256 * i;
    s += (idx < remain) ? src[idx] : 0.f;
  }
  s *= C;
  __shared__ float wsum[8];
  s = waveReduceAdd(s);
  int wave = threadIdx.x >> 5, lane = threadIdx.x & 31;
  if (lane == 0) wsum[wave] = s;
  __syncthreads();
  if (threadIdx.x == 0) {
    float t = 0.f;
#pragma unroll
    for (int i = 0; i < 8; ++i) t += wsum[i];
    partials[(size_t)b * nblk + blk] = t;
  }
}

// ---------------- Pass 2: exclusive scan of block partials (tiny) ----------------
__global__ __launch_bounds__(256) void sg_scan(const float* __restrict__ partials,
                                               float* __restrict__ offsets,
                                               int nblk, int B) {
  int wave = threadIdx.x >> 5, lane = threadIdx.x & 31;
  if (wave < B && lane == 0) {
    float run = 0.f;
    for (int i = 0; i < nblk; ++i) {
      offsets[(size_t)wave * nblk + i] = run;
      run += partials[(size_t)wave * nblk + i];
    }
  }
}

// ---------------- Pass 3: WMMA tile scan + harmonic synth (fused) ----------------
__global__ __launch_bounds__(256) void sg_main(const float* __restrict__ f0,
                                               const float* __restrict__ W,
                                               const float* __restrict__ bias,
                                               const float* __restrict__ blockOff,
                                               float* __restrict__ out,
                                               int N, int nblk) {
  const float C = 6.28318530717958647692f / (float)SRATE;
  __shared__ float lds[16 * TILE_LDS];
  __shared__ float tileSums[16];

  int b = blockIdx.y, blk = blockIdx.x;
  const float* src = f0 + (size_t)b * N + (size_t)blk * CHUNK;
  float* dst = out + (size_t)b * N;
  int remain = N - blk * CHUNK;
  int tid = threadIdx.x;

  // Stage c*f0, zero-padded. Column-major 16x16 tiles: linear ts = col*16 + row,
  // stored at lds[tile*272 + col*17 + row] (pad-17 keeps LDS banks distinct).
#pragma unroll
  for (int i = 0; i < CHUNK / 256; ++i) {
    int s = tid + 256 * i;
    float v = (s < remain) ? src[s] * C : 0.f;
    int t = s >> 8, ts = s & 255;
    lds[t * TILE_LDS + (ts >> 4) * PADCOL + (ts & 15)] = v;
  }
  __syncthreads();

  int wave = tid >> 5, lane = tid & 31;
  int lo = lane & 15;
  bool hi = lane >= 16;

  // A operand = constant lower-triangular ones L (inclusive scan matrix),
  // per ISA 16x4 f32 A layout: lane M=lo; VGPR0 K=4c+(hi?2:0), VGPR1 K+1.
  v2f A[4];
#pragma unroll
  for (int c = 0; c < 4; ++c) {
    int kx = 4 * c + (hi ? 2 : 0);
    A[c][0] = (lo >= kx)     ? 1.f : 0.f;
    A[c][1] = (lo >= kx + 1) ? 1.f : 0.f;
  }

  float Wc[NH];
#pragma unroll
  for (int h = 0; h < NH; ++h) Wc[h] = W[h];
  float bb = bias[0];
  float blockOffset = blockOff[(size_t)b * nblk + blk];

  int t0 = wave * 2;

  // B operands (rows of X) for this wave's two tiles + tile totals.
  v2f Bm[2][4];
  float tsum0 = 0.f;
#pragma unroll
  for (int tt = 0; tt < 2; ++tt) {
    int t = t0 + tt;
    float acc = 0.f;
#pragma unroll
    for (int c = 0; c < 4; ++c) {
      int kx = 4 * c + (hi ? 2 : 0);
      float b0 = lds[t * TILE_LDS + lo * PADCOL + kx];
      float b1 = lds[t * TILE_LDS + lo * PADCOL + kx + 1];
      Bm[tt][c][0] = b0;
      Bm[tt][c][1] = b1;
      acc += b0 + b1;
    }
    acc = waveReduceAdd(acc);
    if (tt == 0) tsum0 = acc;
    if (lane == 0) tileSums[t] = acc;
  }
  __syncthreads();

  // Exclusive intra-block tile offset for first tile of this wave.
  float vsel = (lane < t0) ? tileSums[lane] : 0.f;
  float off0 = waveReduceAdd(vsel) + blockOffset;
  float tileOffs[2] = { off0, off0 + tsum0 };

#pragma unroll
  for (int tt = 0; tt < 2; ++tt) {
    int t = t0 + tt;
    // Y = L @ X : all 16 within-column inclusive scans via 4 chained f32 WMMAs.
    v8f Y = {};
#pragma unroll
    for (int c = 0; c < 4; ++c) {
      Y = __builtin_amdgcn_wmma_f32_16x16x4_f32(false, A[c], false, Bm[tt][c],
                                                (short)0, Y, false, false);
    }
    // Column sums = row M=15 of Y -> component 7, lanes 16..31 (N = lane-16).
    float colsum = __shfl(Y[7], 16 + lo, 32);
    float inc = colsum;
#pragma unroll
    for (int d = 1; d < 16; d <<= 1) {
      float u = __shfl_up(inc, d, 16);
      if (lo >= d) inc += u;
    }
    float base = (inc - colsum) + tileOffs[tt];  // exclusive column offset + carries

    int rbase = hi ? 8 : 0;
    float res[8];
#pragma unroll
    for (int r = 0; r < 8; ++r) {
      float p = Y[r] + base;  // phase P at sample (lo*16 + rbase + r)
      float cf = lds[t * TILE_LDS + lo * PADCOL + rbase + r];
      float voiced = (cf > 10.f * C) ? 1.f : 0.f;
      // 9 harmonics from one sin + one cos via Chebyshev recurrence.
      float sp = __sinf(p);
      float cp = __cosf(p);
      float c2 = 2.f * cp;
      float sm1 = 0.f, sh = sp;
      float a = Wc[0] * sh;
#pragma unroll
      for (int h = 1; h < NH; ++h) {
        float sn = c2 * sh - sm1;
        sm1 = sh;
        sh = sn;
        a += Wc[h] * sh;
      }
      float mixed = 0.1f * voiced * a + bb;
      float e = __expf(-2.f * fabsf(mixed));
      float th = (1.f - e) / (1.f + e);
      res[r] = copysignf(th, mixed);
    }
    // D layout hands each lane 8 contiguous samples -> two b128 stores.
    long gbase = (long)blk * CHUNK + (long)t * TILE + lo * 16 + rbase;
    if (gbase < N) {  // N % 16 == 0 and gbase % 8 == 0 -> all-or-nothing
      float4 r0, r1;
      r0.x = res[0]; r0.y = res[1]; r0.z = res[2]; r0.w = res[3];
      r1.x = res[4]; r1.y = res[5]; r1.z = res[6]; r1.w = res[7];
      *(float4*)(dst + gbase)     = r0;
      *(float4*)(dst + gbase + 4) = r1;
    }
  }
}

extern "C" void kernel_launch(void* const* d_in, const int* in_sizes, int n_in,
                              void* d_out, int out_size, void* d_ws, size_t ws_size,
                              hipStream_t stream) {
  const float* f0 = (const float*)d_in[0];
  const float* W  = (const float*)d_in[1];
  const float* bb = (const float*)d_in[2];
  float* out = (float*)d_out;

  const int N = NSAMP;
  int B = in_sizes[0] / N;
  int nblk = (N + CHUNK - 1) / CHUNK;

  float* partials = (float*)d_ws;
  float* offsets  = partials + (size_t)B * nblk;

  sg_partials<<<dim3(nblk, B), 256, 0, stream>>>(f0, partials, N, nblk);
  sg_scan<<<1, 256, 0, stream>>>(partials, offsets, nblk, B);
  sg_main<<<dim3(nblk, B), 256, 0, stream>>>(f0, W, bb, offsets, out, N, nblk);
}